// GCNSkipBackbone_8770323219001
// MI455X (gfx1250) — compile-verified
//
#include <hip/hip_runtime.h>
#include <hip/hip_bf16.h>
#include <math.h>

#define EPS 1e-5f

typedef float v2f __attribute__((ext_vector_type(2)));
typedef float v8f __attribute__((ext_vector_type(8)));

// ---------------------------------------------------------------------------
// Graph normalization (layer-invariant): deg, dis = rsqrt(deg+1), coef[e]
// ---------------------------------------------------------------------------
__global__ void zero_f32_kernel(float* __restrict__ p, int n) {
    int i = blockIdx.x * blockDim.x + threadIdx.x;
    if (i < n) p[i] = 0.0f;
}

__global__ void deg_accum_kernel(const int* __restrict__ dst, float* __restrict__ deg, int E) {
    int e = blockIdx.x * blockDim.x + threadIdx.x;
    if (e < E) unsafeAtomicAdd(&deg[dst[e]], 1.0f);
}

__global__ void dis_kernel(const float* __restrict__ deg, float* __restrict__ dis, int N) {
    int n = blockIdx.x * blockDim.x + threadIdx.x;
    if (n < N) dis[n] = rsqrtf(deg[n] + 1.0f);
}

__global__ void coef_kernel(const int* __restrict__ src, const int* __restrict__ dst,
                            const float* __restrict__ dis, float* __restrict__ coef, int E) {
    int e = blockIdx.x * blockDim.x + threadIdx.x;
    if (e < E) coef[e] = dis[src[e]] * dis[dst[e]];
}

// ---------------------------------------------------------------------------
// FP32 WMMA GEMM: C[M,Fo] = A[M,K] * W[K,Fo]
// One wave computes a 16(M) x 64(N) strip: one A fragment is reused against
// 4 B fragments per k-step (4x v_wmma_f32_16x16x4_f32 per 5 wave-loads).
// A-frag (16x4 f32): lanes 0-15 hold row M=lane, K=k0+{0,1}; lanes 16-31 K=k0+{2,3}.
// B-frag (4x16 f32): lanes 0-15 hold col N=lane, K=k0+{0,1}; lanes 16-31 K=k0+{2,3}.
// C/D (16x16 f32):  VGPR r: lanes 0-15 -> M=r, lanes 16-31 -> M=8+r; N=lane&15.
// ---------------------------------------------------------------------------
__global__ void __launch_bounds__(256)
gemm_wmma_f32_kernel(const float* __restrict__ A, const float* __restrict__ W,
                     float* __restrict__ C, int M, int K, int Fo) {
    const int lane = threadIdx.x & 31;
    const int wave = threadIdx.x >> 5;
    const int half = lane >> 4;     // 0: lanes 0-15, 1: lanes 16-31
    const int ll   = lane & 15;

    const int m0 = (blockIdx.y * 8 + wave) * 16;   // 16-row tile (wave-uniform)
    const int n0 = blockIdx.x * 64;                // 64-col strip
    if (m0 >= M) return;                           // wave-uniform exit: EXEC all-1 in live waves

    v8f acc0 = {}, acc1 = {}, acc2 = {}, acc3 = {};
    const float* ap = A + (size_t)(m0 + ll) * K + 2 * half;    // row m0+ll
    const float* bp = W + (size_t)(2 * half) * Fo + (n0 + ll); // col n0+ll (+16j per tile)

    for (int k0 = 0; k0 < K; k0 += 4) {
        v2f a;
        a.x = ap[k0];
        a.y = ap[k0 + 1];
        const float* b0 = bp + (size_t)k0 * Fo;        // row k0+2*half of W
        const float* b1 = bp + (size_t)(k0 + 1) * Fo;  // row k0+2*half+1 of W
        v2f b;
        b.x = b0[0];  b.y = b1[0];
        acc0 = __builtin_amdgcn_wmma_f32_16x16x4_f32(false, a, false, b, (short)0, acc0, false, false);
        b.x = b0[16]; b.y = b1[16];
        acc1 = __builtin_amdgcn_wmma_f32_16x16x4_f32(false, a, false, b, (short)0, acc1, false, false);
        b.x = b0[32]; b.y = b1[32];
        acc2 = __builtin_amdgcn_wmma_f32_16x16x4_f32(false, a, false, b, (short)0, acc2, false, false);
        b.x = b0[48]; b.y = b1[48];
        acc3 = __builtin_amdgcn_wmma_f32_16x16x4_f32(false, a, false, b, (short)0, acc3, false, false);
    }

    float* crow = C + (size_t)(m0 + 8 * half) * Fo + (n0 + ll);
#pragma unroll
    for (int r = 0; r < 8; ++r) {
        crow[(size_t)r * Fo +  0] = acc0[r];
        crow[(size_t)r * Fo + 16] = acc1[r];
        crow[(size_t)r * Fo + 32] = acc2[r];
        crow[(size_t)r * Fo + 48] = acc3[r];
    }
}

// ---------------------------------------------------------------------------
// agg[n,c] = hw[n,c] * dis[n]^2   (self-loop term; also initializes agg)
// ---------------------------------------------------------------------------
__global__ void init_agg_kernel(const float* __restrict__ hw, const float* __restrict__ dis,
                                float* __restrict__ agg, int N) {
    int t = blockIdx.x * blockDim.x + threadIdx.x;   // N * 64 threads (4 ch each)
    int n = t >> 6;
    int c = (t & 63) * 4;
    if (n >= N) return;
    float d = dis[n];
    float d2 = d * d;
    float4 v = *(const float4*)(hw + (size_t)n * 256 + c);
    v.x *= d2; v.y *= d2; v.z *= d2; v.w *= d2;
    *(float4*)(agg + (size_t)n * 256 + c) = v;
}

// ---------------------------------------------------------------------------
// Edge scatter: agg[dst] += hw[src] * coef[e].  hw/agg are L2-resident (51MB
// each vs 192MB L2) so the gather + global_atomic_add_f32 stay on-chip.
// 64 lanes per edge, float4 per lane.
// ---------------------------------------------------------------------------
__global__ void scatter_kernel(const float* __restrict__ hw, const float* __restrict__ coef,
                               const int* __restrict__ src, const int* __restrict__ dst,
                               float* __restrict__ agg, int E) {
    long long t = (long long)blockIdx.x * blockDim.x + threadIdx.x;
    int e = (int)(t >> 6);
    int c = ((int)t & 63) * 4;
    if (e >= E) return;
    float cf = coef[e];
    float4 v = *(const float4*)(hw + (size_t)src[e] * 256 + c);
    float* out = agg + (size_t)dst[e] * 256 + c;
    unsafeAtomicAdd(out + 0, v.x * cf);
    unsafeAtomicAdd(out + 1, v.y * cf);
    unsafeAtomicAdd(out + 2, v.z * cf);
    unsafeAtomicAdd(out + 3, v.w * cf);
}

// ---------------------------------------------------------------------------
// Fused epilogue: +bias, nan_to_num, LayerNorm(g,be), optional skip, ReLU.
// One wave32 per node (256 channels -> 8 per lane), shuffle reductions.
// ---------------------------------------------------------------------------
__global__ void __launch_bounds__(256)
post_kernel(const float* __restrict__ agg, const float* __restrict__ bias,
            const float* __restrict__ gamma, const float* __restrict__ beta,
            const float* __restrict__ hprev, float* __restrict__ hout,
            int N, int do_skip, int do_relu) {
    const int C = 256;
    int wave = threadIdx.x >> 5;
    int lane = threadIdx.x & 31;
    int n = blockIdx.x * (blockDim.x >> 5) + wave;
    if (n >= N) return;

    const float* arow = agg + (size_t)n * C;
    float v[8];
    float s = 0.0f;
#pragma unroll
    for (int i = 0; i < 8; ++i) {
        int c = lane + 32 * i;
        float x = arow[c] + bias[c];
        if (isnan(x))      x = 0.0f;
        else if (isinf(x)) x = (x > 0.0f) ? EPS : -EPS;
        v[i] = x;
        s += x;
    }
#pragma unroll
    for (int off = 16; off > 0; off >>= 1) s += __shfl_xor(s, off, 32);
    float mu = s * (1.0f / C);

    float var = 0.0f;
#pragma unroll
    for (int i = 0; i < 8; ++i) {
        v[i] -= mu;
        var += v[i] * v[i];
    }
#pragma unroll
    for (int off = 16; off > 0; off >>= 1) var += __shfl_xor(var, off, 32);
    var *= (1.0f / C);
    float inv = rsqrtf(var + EPS);

    float* orow = hout + (size_t)n * C;
    const float* prow = hprev ? hprev + (size_t)n * C : nullptr;
#pragma unroll
    for (int i = 0; i < 8; ++i) {
        int c = lane + 32 * i;
        float y = v[i] * inv * gamma[c] + beta[c];
        if (do_skip) y += prow[c];
        if (do_relu) y = fmaxf(y, 0.0f);
        orow[c] = y;
    }
}

// ---------------------------------------------------------------------------
// Host-side orchestration
// ---------------------------------------------------------------------------
extern "C" void kernel_launch(void* const* d_in, const int* in_sizes, int n_in,
                              void* d_out, int out_size, void* d_ws, size_t ws_size,
                              hipStream_t stream) {
    const int IN_C = 128, FO = 256;
    const int N = in_sizes[0] / IN_C;      // 50000
    const int E = in_sizes[1] / 2;         // 1600000

    const float* x   = (const float*)d_in[0];
    const int*   src = (const int*)d_in[1];
    const int*   dst = src + E;
    const float* W[4], *b[4], *g[4], *be[4];
    for (int i = 0; i < 4; ++i) {
        W[i]  = (const float*)d_in[2 + 4 * i + 0];
        b[i]  = (const float*)d_in[2 + 4 * i + 1];
        g[i]  = (const float*)d_in[2 + 4 * i + 2];
        be[i] = (const float*)d_in[2 + 4 * i + 3];
    }

    float* ws = (float*)d_ws;
    const size_t NC = (size_t)N * FO;
    float* bufA = ws;                    // hw / h ping-pong
    float* bufB = ws + NC;
    float* agg  = ws + 2 * NC;
    float* deg  = ws + 3 * NC;
    float* dis  = deg + N;
    float* coef = dis + N;

    const int B = 256;
    // --- graph normalization (once) ---
    zero_f32_kernel<<<(N + B - 1) / B, B, 0, stream>>>(deg, N);
    deg_accum_kernel<<<(E + B - 1) / B, B, 0, stream>>>(dst, deg, E);
    dis_kernel<<<(N + B - 1) / B, B, 0, stream>>>(deg, dis, N);
    coef_kernel<<<(E + B - 1) / B, B, 0, stream>>>(src, dst, dis, coef, E);

    const int rowTiles  = (N + 15) / 16;                 // 3125
    const dim3 gemmGrid(FO / 64, (rowTiles + 7) / 8);    // 4 x 391
    const int aggBlocks  = (int)((NC / 4 + B - 1) / B);  // N*64 threads
    const int scatBlocks = (int)(((long long)E * 64 + B - 1) / B);
    const int postBlocks = (N + 7) / 8;                  // 8 nodes / 256-thr block

    // Layer schedule: (h_in, K, hw_buf, h_out, skip_src)
    struct { const float* hin; int K; float* hw; float* hout; const float* skip; int relu; }
    L[4] = {
        { x,    IN_C, bufA, bufB,          nullptr, 1 },
        { bufB, FO,   bufA, bufA,          bufB,    1 },   // hout overwrites hw (hw dead by then)
        { bufA, FO,   bufB, bufB,          bufA,    1 },
        { bufB, FO,   bufA, (float*)d_out, bufB,    0 },
    };

    for (int i = 0; i < 4; ++i) {
        gemm_wmma_f32_kernel<<<gemmGrid, B, 0, stream>>>(L[i].hin, W[i], L[i].hw, N, L[i].K, FO);
        init_agg_kernel<<<aggBlocks, B, 0, stream>>>(L[i].hw, dis, agg, N);
        scatter_kernel<<<scatBlocks, B, 0, stream>>>(L[i].hw, coef, src, dst, agg, E);
        post_kernel<<<postBlocks, B, 0, stream>>>(agg, b[i], g[i], be[i],
                                                  L[i].skip, L[i].hout,
                                                  N, L[i].skip != nullptr, L[i].relu);
    }
}